// IoUScore_15504831938841
// MI455X (gfx1250) — compile-verified
//
#include <hip/hip_runtime.h>

// IoU score for (8, 21, 512, 512) fp32 logits vs (8, 512, 512) int32 labels.
// Memory-bound streaming reduction: ~184 MB read once -> ~7.9 us floor at 23.3 TB/s.
// argmax(softmax(x)) == argmax(x), so we skip the softmax entirely.

typedef float v4f __attribute__((ext_vector_type(4)));
typedef int   v4i __attribute__((ext_vector_type(4)));

#define NCLS     21
#define PLANE    262144            // 512*512
#define NPIX     2097152           // 8 * PLANE
#define NGRP     (NPIX / 4)        // float4 pixel groups
#define HSTRIDE  96                // 3 histograms x 32 padded bins
#define REPLICAS 32                // replicated global histogram sets (contention spreading)
#define SMOOTHF  1e-6f

__global__ void iou_init_kernel(unsigned* __restrict__ hist) {
    int i = blockIdx.x * blockDim.x + threadIdx.x;
    if (i < REPLICAS * HSTRIDE) hist[i] = 0u;
}

__global__ __launch_bounds__(256)
void iou_hist_kernel(const float* __restrict__ pred,
                     const int*   __restrict__ tgt,
                     unsigned*    __restrict__ hist) {
    __shared__ unsigned h[HSTRIDE];   // [0,32): cnt_pred  [32,64): cnt_tgt  [64,96): inter
    for (int i = threadIdx.x; i < HSTRIDE; i += blockDim.x) h[i] = 0u;
    __syncthreads();

    const int nt = gridDim.x * blockDim.x;
    for (int g = blockIdx.x * blockDim.x + threadIdx.x; g < NGRP; g += nt) {
        const int pix = g << 2;               // first of 4 consecutive pixels
        const int b   = pix >> 18;            // / PLANE
        const int p   = pix & (PLANE - 1);    // % PLANE (PLANE divisible by 4 -> same batch)
        const float* base = pred + (size_t)b * (size_t)(NCLS * PLANE) + (size_t)p;

        // gfx1250 global_prefetch_b8: pull next grid-stride tile toward the caches
        if (g + nt < NGRP) __builtin_prefetch(base + (size_t)nt * 4, 0, 0);

        // streaming one-shot data: non-temporal b128 loads
        v4f best = __builtin_nontemporal_load((const v4f*)base);
        v4i bi   = {0, 0, 0, 0};
#pragma unroll
        for (int c = 1; c < NCLS; ++c) {
            v4f v = __builtin_nontemporal_load((const v4f*)(base + (size_t)c * PLANE));
#pragma unroll
            for (int k = 0; k < 4; ++k) {
                // strict '>' scanning ascending c == jnp.argmax first-max semantics
                if (v[k] > best[k]) { best[k] = v[k]; bi[k] = c; }
            }
        }

        v4i t = __builtin_nontemporal_load((const v4i*)(tgt + pix));
#pragma unroll
        for (int k = 0; k < 4; ++k) {
            const int pc = bi[k];
            const int tc = t[k];
            atomicAdd(&h[pc], 1u);            // ds_add_u32
            atomicAdd(&h[32 + tc], 1u);
            if (pc == tc) atomicAdd(&h[64 + pc], 1u);
        }
    }
    __syncthreads();

    // flush block-local histogram into one of REPLICAS global sets
    unsigned* dst = hist + (size_t)(blockIdx.x & (REPLICAS - 1)) * HSTRIDE;
    for (int i = threadIdx.x; i < HSTRIDE; i += blockDim.x) {
        const unsigned v = h[i];
        if (v) atomicAdd(&dst[i], v);
    }
}

__global__ void iou_finalize_kernel(const unsigned* __restrict__ hist,
                                    float* __restrict__ out) {
    const int c = threadIdx.x;                // one wave32
    float iou = 0.0f;
    if (c < NCLS) {
        unsigned cp = 0, ct = 0, ci = 0;
        for (int r = 0; r < REPLICAS; ++r) {
            const unsigned* hr = hist + r * HSTRIDE;
            cp += hr[c];
            ct += hr[32 + c];
            ci += hr[64 + c];
        }
        const float inter = (float)ci;
        const float uni   = (float)cp + (float)ct - inter;
        iou = (inter + SMOOTHF) / (uni + SMOOTHF);
    }
    // 32-lane shuffle tree sum (wave32 on gfx1250)
    for (int off = 16; off > 0; off >>= 1) iou += __shfl_down(iou, off, 32);
    if (c == 0) out[0] = iou / (float)NCLS;
}

extern "C" void kernel_launch(void* const* d_in, const int* in_sizes, int n_in,
                              void* d_out, int out_size, void* d_ws, size_t ws_size,
                              hipStream_t stream) {
    const float* pred = (const float*)d_in[0];   // (8, 21, 512, 512) fp32
    const int*   tgt  = (const int*)d_in[1];     // (8, 512, 512) int32
    unsigned*    hist = (unsigned*)d_ws;         // REPLICAS * HSTRIDE u32 = 12 KB
    float*       out  = (float*)d_out;           // scalar fp32

    iou_init_kernel<<<(REPLICAS * HSTRIDE + 255) / 256, 256, 0, stream>>>(hist);
    iou_hist_kernel<<<1024, 256, 0, stream>>>(pred, tgt, hist);
    iou_finalize_kernel<<<1, 32, 0, stream>>>(hist, out);
}